// GeneInteractionNN_72636486909960
// MI455X (gfx1250) — compile-verified
//
#include <hip/hip_runtime.h>
#include <hip/hip_bf16.h>

// Problem constants (match the JAX reference)
#define BB   16384   // batch
#define GG   128     // genes
#define SMAX 64      // max SNPs per gene (K of grouped GEMM)
#define HH   10      // gene-MLP hidden (padded to 16 in WMMA N)
#define PH   100     // predictor hidden (padded to 112 = 7 x 16)

#define S1_STRIDE 68   // 64 + 4 pad: 32-lane b64 reads hit 64 distinct LDS banks
#define S2_STRIDE 132  // 128 + 4 pad: same property

typedef __attribute__((ext_vector_type(2))) float v2f;
typedef __attribute__((ext_vector_type(8))) float v8f;

// ---------------------------------------------------------------------------
// Stage 1: h = relu(x[b,g,:] @ W1[g] + b1[g]);  gl[b,g] = h . W2[g] + b2[g]
// One wave computes a 16(B) x 16(H-padded) tile for one gene via 16 chained
// V_WMMA_F32_16X16X4_F32 (K = 64 = 16 * 4), fp32 end to end.
// W1[g] is staged once per block into LDS, transposed [n][k] and zero-padded,
// so the hot loop's B fragment is one unconditional ds_load_b64 per WMMA.
// x is a 512 MB read-once stream -> non-temporal loads keep the 192 MB L2
// free for weights and the stage-1 -> stage-2 intermediate.
//
// A (16x4 f32) layout: lanes 0-15 -> M=0..15 / K={0,1}; lanes 16-31 -> K={2,3}
// C/D (16x16 f32):     VGPR r: lanes 0-15 -> M=r, lanes 16-31 -> M=r+8; N=lane%16
// ---------------------------------------------------------------------------
__global__ void __launch_bounds__(256)
gene_mlp_wmma_kernel(const float* __restrict__ x,    // [B,G,SMAX]
                     const float* __restrict__ W1,   // [G,SMAX,H]
                     const float* __restrict__ b1,   // [G,H]
                     const float* __restrict__ W2,   // [G,H]
                     const float* __restrict__ b2,   // [G]
                     float* __restrict__ gl)         // [B,G]  (workspace)
{
    __shared__ float w1T[16 * S1_STRIDE];            // [n][k], zero-padded n>=10

    const int tid  = threadIdx.x;
    const int lane = tid & 31;
    const int wid  = tid >> 5;                       // 8 waves / block
    const int g    = blockIdx.x >> 7;                // 128 tile-groups per gene
    const int b0   = (((blockIdx.x & 127) << 3) + wid) << 4;  // 16-row tile base

    const float* __restrict__ w1g = W1 + (size_t)g * SMAX * HH;

    // Cooperative fill: 1024 elements, 4 per thread. Index-clamp keeps the
    // load unconditional (no EXEC-masked branches); pad rows get 0.
    #pragma unroll
    for (int i = 0; i < 4; ++i) {
        const int e = tid + i * 256;                 // 0..1023
        const int n = e >> 6;                        // 0..15
        const int k = e & 63;                        // 0..63
        const int nc = n < HH ? n : HH - 1;
        float v = w1g[(size_t)k * HH + nc];          // W1[g, k, nc]
        w1T[n * S1_STRIDE + k] = (n < HH) ? v : 0.0f;
    }
    __syncthreads();

    const int half = lane >> 4;                      // 0 | 1
    const int mn   = lane & 15;                      // A row / B,C,D column

    // Lane's A row: contiguous 64 floats, read as aligned float2 (global_load_b64)
    const v2f* __restrict__ xrow =
        (const v2f*)(x + ((size_t)(b0 + mn) * GG + g) * SMAX);
    // Lane's B row in LDS: w1T[mn][k0..k0+1] is one aligned ds_load_b64
    const v2f* __restrict__ brow = (const v2f*)(&w1T[mn * S1_STRIDE]);

    v8f acc = {};
    #pragma unroll
    for (int kk = 0; kk < 16; ++kk) {
        // NT hint: x is consumed exactly once, do not cache it in L2
        v2f a = __builtin_nontemporal_load(&xrow[2 * kk + half]);
        v2f b = brow[2 * kk + half];                 // W1T[mn, 4kk+2h : +2]
        acc = __builtin_amdgcn_wmma_f32_16x16x4_f32(
            false, a, false, b, (short)0, acc, false, false);
    }

    // Epilogue: + b1, ReLU, dot with W2 across the N (=H) dimension.
    // Clamped-index loads avoid EXEC-masked conditional loads; w2n = 0 on the
    // padded columns kills their (zero anyway) contribution.
    const int   mnc = mn < HH ? mn : HH - 1;
    const float b1n = b1[g * HH + mnc];
    float       w2n = W2[g * HH + mnc];
    w2n = (mn < HH) ? w2n : 0.0f;

    float partial[8];
    #pragma unroll
    for (int r = 0; r < 8; ++r) {
        float h = acc[r] + b1n;
        h = h > 0.0f ? h : 0.0f;
        partial[r] = h * w2n;
    }
    // Reduce over the 16 N-lanes of each half-wave (wave32 butterfly)
    #pragma unroll
    for (int r = 0; r < 8; ++r) {
        float s = partial[r];
        s += __shfl_xor(s, 1, 32);
        s += __shfl_xor(s, 2, 32);
        s += __shfl_xor(s, 4, 32);
        s += __shfl_xor(s, 8, 32);
        partial[r] = s;
    }
    if (mn == 0) {
        const float bias = b2[g];
        #pragma unroll
        for (int r = 0; r < 8; ++r) {
            const int row = r + 8 * half;            // lanes>=16 hold M=r+8
            // Regular-temporal store: stage 2 re-reads gl from L2 right away
            gl[(size_t)(b0 + row) * GG + g] = partial[r] + bias;
        }
    }
}

// ---------------------------------------------------------------------------
// Stage 2: p = relu(gl @ Wp1 + bp1);  out = p @ Wp2 + bp2
// One wave per 16-row B tile. A (16x128) kept resident in 64 VGPRs; Wp1 is
// staged transposed+padded in LDS ([112][132] floats, ~58 KB of the 320 KB
// WGP pool); 7 N-tiles x 32 chained fp32 WMMAs (K = 128).
// ---------------------------------------------------------------------------
__global__ void __launch_bounds__(256)
predictor_wmma_kernel(const float* __restrict__ gl,   // [B,G]
                      const float* __restrict__ Wp1,  // [G,PH]
                      const float* __restrict__ bp1,  // [PH]
                      const float* __restrict__ Wp2,  // [PH]
                      const float* __restrict__ bp2,  // [1]
                      float* __restrict__ out)        // [B]
{
    __shared__ float wpT[112 * S2_STRIDE];           // [col][k], cols>=100 zero

    const int tid  = threadIdx.x;
    const int lane = tid & 31;
    const int wid  = tid >> 5;
    const int b0   = (blockIdx.x * 8 + wid) << 4;

    // Cooperative fill: 112*128 = 14336 elements, 56 per thread.
    #pragma unroll
    for (int i = 0; i < 56; ++i) {
        const int e = tid + i * 256;                 // 0..14335
        const int c = e >> 7;                        // col 0..111
        const int k = e & 127;                       // k   0..127
        const int cc = c < PH ? c : PH - 1;
        float v = Wp1[(size_t)k * PH + cc];          // Wp1[k, cc]
        wpT[c * S2_STRIDE + k] = (c < PH) ? v : 0.0f;
    }
    __syncthreads();

    const int half = lane >> 4;
    const int mn   = lane & 15;

    // Resident A: this lane's 128-float gene_layer row, as 32 aligned float2s
    const v2f* __restrict__ arow = (const v2f*)(gl + (size_t)(b0 + mn) * GG);
    v2f a[32];
    #pragma unroll
    for (int kk = 0; kk < 32; ++kk) a[kk] = arow[2 * kk + half];

    float result[8];
    #pragma unroll
    for (int r = 0; r < 8; ++r) result[r] = 0.0f;

    for (int nt = 0; nt < 7; ++nt) {                 // 7 * 16 = 112 >= PH
        const int col = nt * 16 + mn;
        const v2f* __restrict__ brow = (const v2f*)(&wpT[col * S2_STRIDE]);

        v8f acc = {};
        #pragma unroll
        for (int kk = 0; kk < 32; ++kk) {
            v2f b = brow[2 * kk + half];             // one ds_load_b64
            acc = __builtin_amdgcn_wmma_f32_16x16x4_f32(
                false, a[kk], false, b, (short)0, acc, false, false);
        }

        // acc on padded cols is exactly 0 (B cols zeroed), so zeroing w2
        // suffices; bias load uses a clamped index to stay unconditional.
        const int   colc = col < PH ? col : PH - 1;
        const float bias = bp1[colc];
        float       w2   = Wp2[colc];
        w2 = (col < PH) ? w2 : 0.0f;

        #pragma unroll
        for (int r = 0; r < 8; ++r) {
            float p = acc[r] + bias;
            p = p > 0.0f ? p : 0.0f;
            result[r] += p * w2;                     // fold PH->1 projection in
        }
    }

    #pragma unroll
    for (int r = 0; r < 8; ++r) {
        float s = result[r];
        s += __shfl_xor(s, 1, 32);
        s += __shfl_xor(s, 2, 32);
        s += __shfl_xor(s, 4, 32);
        s += __shfl_xor(s, 8, 32);
        result[r] = s;
    }
    if (mn == 0) {
        const float bias = bp2[0];
        #pragma unroll
        for (int r = 0; r < 8; ++r) {
            out[b0 + r + 8 * half] = result[r] + bias;
        }
    }
}

extern "C" void kernel_launch(void* const* d_in, const int* in_sizes, int n_in,
                              void* d_out, int out_size, void* d_ws, size_t ws_size,
                              hipStream_t stream) {
    const float* x   = (const float*)d_in[0];
    const float* W1  = (const float*)d_in[1];
    const float* b1  = (const float*)d_in[2];
    const float* W2  = (const float*)d_in[3];
    const float* b2  = (const float*)d_in[4];
    const float* Wp1 = (const float*)d_in[5];
    const float* bp1 = (const float*)d_in[6];
    const float* Wp2 = (const float*)d_in[7];
    const float* bp2 = (const float*)d_in[8];
    float* out = (float*)d_out;

    float* gl = (float*)d_ws;                        // [B,G] fp32 = 8 MB scratch

    // Stage 1: 128 genes x 128 tile-groups; 8 waves/block, one gene per block
    dim3 grid1(GG * (BB / 16 / 8));                  // 16384 blocks
    gene_mlp_wmma_kernel<<<grid1, 256, 0, stream>>>(x, W1, b1, W2, b2, gl);

    // Stage 2: 1024 B-tiles, 8 waves/block
    dim3 grid2(BB / 16 / 8);                         // 128 blocks
    predictor_wmma_kernel<<<grid2, 256, 0, stream>>>(gl, Wp1, bp1, Wp2, bp2, out);
}